// Attention_4011499454782
// MI455X (gfx1250) — compile-verified
//
#include <hip/hip_runtime.h>
#include <hip/hip_bf16.h>
#include <math.h>

// ---- problem constants (match reference) ----
#define NUM_SEQS   32
#define KV_LEN     2048
#define NUM_HEADS  16
#define HEAD_SIZE  128
#define TOTAL_SLOTS (NUM_SEQS * KV_LEN)
#define SCALE      0.08838834764831845f

#define WAVES      4
#define THREADS    (WAVES * 32)
#define TOK_PER_WAVE (KV_LEN / WAVES)   // 512

typedef __attribute__((ext_vector_type(2))) float v2f;
typedef __attribute__((ext_vector_type(8))) float v8f;

// Per-(seq,head) decode attention. Memory-bound: ~2 MB of K+V per block,
// 512 blocks -> 1.07 GB total @ 23.3 TB/s ~= 46 us floor.
__global__ __launch_bounds__(THREADS)
void paged_attn_decode(const float* __restrict__ q,
                       const float* __restrict__ k_new,
                       const float* __restrict__ v_new,
                       const float* __restrict__ kv_cache,
                       const long long* __restrict__ slot_mapping,
                       float* __restrict__ out) {
    __shared__ float lds_score[KV_LEN];     // scores, then probs (8 KB)
    __shared__ float lds_q[HEAD_SIZE];
    __shared__ float lds_out[HEAD_SIZE];
    __shared__ float red[WAVES];
    __shared__ float stat[2];               // [0]=max, [1]=sum

    const int s    = (int)blockIdx.x >> 4;   // / NUM_HEADS
    const int h    = (int)blockIdx.x & 15;   // % NUM_HEADS
    const int tid  = (int)threadIdx.x;
    const int wave = tid >> 5;
    const int lane = tid & 31;
    const int n    = lane & 15;              // token-in-tile / N column
    const int half = lane >> 4;              // lane half selects k-pair

    const long long row_stride = (long long)NUM_HEADS * HEAD_SIZE; // 2048 floats
    const float* Kc = kv_cache + (long long)s * KV_LEN * row_stride + (long long)h * HEAD_SIZE;
    const float* Vc = Kc + (long long)TOTAL_SLOTS * row_stride;    // V plane
    const float* qv = q     + (long long)(s * NUM_HEADS + h) * HEAD_SIZE;
    const float* kn = k_new + (long long)(s * NUM_HEADS + h) * HEAD_SIZE;
    const float* vn = v_new + (long long)(s * NUM_HEADS + h) * HEAD_SIZE;

    // stage q; zero output accumulator (THREADS == HEAD_SIZE)
    lds_q[tid]   = qv[tid];
    lds_out[tid] = 0.0f;
    __syncthreads();

    const int t_ov = (int)(slot_mapping[s] - (long long)s * KV_LEN); // overridden token

    // ---------------- QK^T via V_WMMA_F32_16X16X4_F32 ----------------
    // C[m,n] += sum_k A[m,k]*B[k,n];  A = q chunk broadcast over M rows,
    // B = K tile (k = dim within 4-chunk, n = token). After 32 chunks,
    // C[*, n] = score(token t0+n), duplicated across M and lane halves.
    const int tbase = wave * TOK_PER_WAVE;
    for (int t0 = tbase; t0 < tbase + TOK_PER_WAVE; t0 += 16) {
        v8f acc = {0.f, 0.f, 0.f, 0.f, 0.f, 0.f, 0.f, 0.f};
        const float* Krow = Kc + (long long)(t0 + n) * row_stride;
        #pragma unroll 8
        for (int c4 = 0; c4 < HEAD_SIZE; c4 += 4) {
            const int d = c4 + 2 * half;                 // this lane's k-pair dims
            v2f a; a.x = lds_q[d];  a.y = lds_q[d + 1];  // A: q broadcast (LDS bcast)
            const float2 kk = *(const float2*)(Krow + d);
            v2f b; b.x = kk.x;      b.y = kk.y;          // B: K tile column
            acc = __builtin_amdgcn_wmma_f32_16x16x4_f32(
                false, a, false, b, (short)0, acc, false, false);
        }
        if (half == 0) lds_score[t0 + n] = acc[0] * SCALE;
    }
    __syncthreads();

    // Fresh-token score: dot(q, k_new) replaces the stale cached row's score.
    if (wave == 0) {
        const int d = lane * 4;
        const float4 kk = *(const float4*)(kn + d);
        const float4 qq = *(const float4*)(qv + d);
        float part = qq.x * kk.x + qq.y * kk.y + qq.z * kk.z + qq.w * kk.w;
        #pragma unroll
        for (int m = 16; m >= 1; m >>= 1) part += __shfl_xor(part, m, 32);
        if (lane == 0) lds_score[t_ov] = part * SCALE;
    }
    __syncthreads();

    // ---------------- softmax over 2048 scores in LDS ----------------
    float mloc = -3.0e38f;
    #pragma unroll
    for (int t = tid; t < KV_LEN; t += THREADS) mloc = fmaxf(mloc, lds_score[t]);
    #pragma unroll
    for (int m = 16; m >= 1; m >>= 1) mloc = fmaxf(mloc, __shfl_xor(mloc, m, 32));
    if (lane == 0) red[wave] = mloc;
    __syncthreads();
    if (tid == 0) {
        float m = red[0];
        for (int w = 1; w < WAVES; ++w) m = fmaxf(m, red[w]);
        stat[0] = m;
    }
    __syncthreads();
    const float mglob = stat[0];

    float lloc = 0.0f;
    #pragma unroll
    for (int t = tid; t < KV_LEN; t += THREADS) {
        const float p = __expf(lds_score[t] - mglob);
        lds_score[t] = p;                       // scores -> probs in place
        lloc += p;
    }
    #pragma unroll
    for (int m = 16; m >= 1; m >>= 1) lloc += __shfl_xor(lloc, m, 32);
    if (lane == 0) red[wave] = lloc;
    __syncthreads();
    if (tid == 0) {
        float l = red[0];
        for (int w = 1; w < WAVES; ++w) l += red[w];
        stat[1] = l;
    }
    __syncthreads();
    const float inv_l = 1.0f / stat[1];

    // ---------------- PV: coalesced float4 V loads, VALU FMA ----------------
    // Each lane owns 4 head dims; 32 lanes cover one contiguous 512 B row.
    const int d = lane * 4;
    float4 acc4 = make_float4(0.f, 0.f, 0.f, 0.f);
    #pragma unroll 4
    for (int t = tbase; t < tbase + TOK_PER_WAVE; ++t) {
        const float* vr = (t == t_ov) ? vn : (Vc + (long long)t * row_stride);
        const float  p  = lds_score[t];         // LDS broadcast
        const float4 vv = *(const float4*)(vr + d);
        acc4.x += p * vv.x;  acc4.y += p * vv.y;
        acc4.z += p * vv.z;  acc4.w += p * vv.w;
    }
    atomicAdd(&lds_out[d + 0], acc4.x);
    atomicAdd(&lds_out[d + 1], acc4.y);
    atomicAdd(&lds_out[d + 2], acc4.z);
    atomicAdd(&lds_out[d + 3], acc4.w);
    __syncthreads();

    out[(long long)(s * NUM_HEADS + h) * HEAD_SIZE + tid] = lds_out[tid] * inv_l;
}

extern "C" void kernel_launch(void* const* d_in, const int* in_sizes, int n_in,
                              void* d_out, int out_size, void* d_ws, size_t ws_size,
                              hipStream_t stream) {
    const float*     q_p   = (const float*)d_in[0];
    const float*     k_p   = (const float*)d_in[1];
    const float*     v_p   = (const float*)d_in[2];
    const float*     kc_p  = (const float*)d_in[3];
    const long long* sm_p  = (const long long*)d_in[4];
    float*           out_p = (float*)d_out;

    dim3 grid(NUM_SEQS * NUM_HEADS);   // 512 blocks: one per (seq, head)
    dim3 block(THREADS);               // 4 wave32s
    paged_attn_decode<<<grid, block, 0, stream>>>(q_p, k_p, v_p, kc_p, sm_p, out_p);
}